// AUTKCLoss_54717883351223
// MI455X (gfx1250) — compile-verified
//
#include <hip/hip_runtime.h>
#include <hip/hip_bf16.h>
#include <stdint.h>

#define WG     256
#define DEPTH  8          // async pipeline depth (tiles in flight)
#define TILE   (WG * 4)   // 1024 floats per tile (4 per thread)
#define KTOP   6          // K+1 negatives kept

#if defined(__has_builtin)
#if __has_builtin(__builtin_amdgcn_global_load_async_to_lds_b32)
#define HAVE_ASYNC_LDS 1
#endif
#if __has_builtin(__builtin_amdgcn_s_wait_asynccnt)
#define HAVE_WAIT_ASYNC 1
#endif
#endif

typedef __attribute__((address_space(3))) int lds_int;
typedef __attribute__((address_space(1))) int glb_int;

// generic LDS pointer's low 32 bits are the LDS offset
static __device__ __forceinline__ lds_int* to_lds(void* p) {
  return (lds_int*)(uint32_t)(uintptr_t)p;
}
static __device__ __forceinline__ glb_int* to_glb(const void* p) {
  return (glb_int*)(uintptr_t)p;
}

#if defined(HAVE_WAIT_ASYNC)
#define WAIT_ASYNC_IMM(n) __builtin_amdgcn_s_wait_asynccnt(n)
#else
#define WAIT_ASYNC_IMM(n) asm volatile("s_wait_asynccnt %0" ::"i"(n) : "memory")
#endif

// wait until at most `remain_tiles` tiles (4 async ops each) are outstanding
static __device__ __forceinline__ void wait_async_tiles(int remain_tiles) {
#if defined(HAVE_ASYNC_LDS)
  switch (remain_tiles) {
    case 7:  WAIT_ASYNC_IMM(28); break;
    case 6:  WAIT_ASYNC_IMM(24); break;
    case 5:  WAIT_ASYNC_IMM(20); break;
    case 4:  WAIT_ASYNC_IMM(16); break;
    case 3:  WAIT_ASYNC_IMM(12); break;
    case 2:  WAIT_ASYNC_IMM(8);  break;
    case 1:  WAIT_ASYNC_IMM(4);  break;
    default: WAIT_ASYNC_IMM(0);  break;
  }
#else
  (void)remain_tiles;
#endif
}

static __device__ __forceinline__ void top6_insert(float (&t)[KTOP], float x) {
  int mi = 0;
  float mn = t[0];
#pragma unroll
  for (int i = 1; i < KTOP; ++i) {
    if (t[i] < mn) { mn = t[i]; mi = i; }
  }
  if (x > mn) t[mi] = x;
}

static __device__ __forceinline__ float top6_min(const float (&t)[KTOP]) {
  float mn = t[0];
#pragma unroll
  for (int i = 1; i < KTOP; ++i) mn = fminf(mn, t[i]);
  return mn;
}

__global__ __launch_bounds__(WG) void autkc_row_kernel(
    const float* __restrict__ pred, const int* __restrict__ y,
    float* __restrict__ row_loss, int C) {
  __shared__ __align__(16) float tileBuf[DEPTH * TILE];  // 32 KB staging
  __shared__ float red[8 * 9];                           // 8 waves x {m,s,xy,t0..5}

  const int row = blockIdx.x;
  const int tid = threadIdx.x;
  const float* rp = pred + (size_t)row * (size_t)C;
  const int yv = y[row];

  float m  = -__builtin_inff();  // running max logit
  float s  = 0.0f;               // running sum exp(x - m)
  float xy = -__builtin_inff();  // logit at the positive class
  float t[KTOP];
#pragma unroll
  for (int i = 0; i < KTOP; ++i) t[i] = -__builtin_inff();
  float tmin = -__builtin_inff();

  auto consume = [&](float x, int col) {
    // online softmax accumulator: one v_exp_f32 per element
    if (x > m) {
      s = s * __expf(m - x) + 1.0f;
      m = x;
    } else {
      s += __expf(x - m);
    }
    if (col == yv) {
      xy = x;                      // exclude positive from top-k
    } else if (x > tmin) {
      top6_insert(t, x);
      tmin = top6_min(t);
    }
  };

  const int numTiles = C / TILE;

#if defined(HAVE_ASYNC_LDS)
  // ---- CDNA5 path: async global->LDS DMA, 8-deep pipeline (ASYNCcnt) ----
  auto issue = [&](int tileIdx) {
    const int stage = tileIdx % DEPTH;
    const float* g = rp + tileIdx * TILE + tid;       // coalesced across lanes
    char* lbase = (char*)&tileBuf[stage * TILE] + tid * 16;
#pragma unroll
    for (int k = 0; k < 4; ++k) {
      // 4-byte granule: rows of 50257 floats are only 4B-aligned
      __builtin_amdgcn_global_load_async_to_lds_b32(
          to_glb(g + k * WG), to_lds((void*)(lbase + k * 4)), 0, 0);
    }
  };

  const int pre = numTiles < DEPTH ? numTiles : DEPTH;
  for (int i = 0; i < pre; ++i) issue(i);

  for (int tIdx = 0; tIdx < numTiles; ++tIdx) {
    int remain = numTiles - 1 - tIdx;
    if (remain > DEPTH - 1) remain = DEPTH - 1;
    wait_async_tiles(remain);                 // oldest tile's 4 async ops done
    asm volatile("" ::: "memory");

    const int stage = tIdx % DEPTH;
    const float4 v =
        *(const float4*)((const char*)&tileBuf[stage * TILE] + tid * 16);
    const int base = tIdx * TILE + tid;
    consume(v.x, base);
    consume(v.y, base + WG);
    consume(v.z, base + 2 * WG);
    consume(v.w, base + 3 * WG);

    const int nxt = tIdx + DEPTH;
    if (nxt < numTiles) {
      // make sure our ds_load of this stage retired before the DMA rewrites it
      asm volatile("s_wait_dscnt 0" ::: "memory");
      issue(nxt);
    }
  }
#else
  // ---- fallback: direct global loads ----
  for (int tIdx = 0; tIdx < numTiles; ++tIdx) {
    const int base = tIdx * TILE + tid;
#pragma unroll
    for (int k = 0; k < 4; ++k) consume(rp[base + k * WG], base + k * WG);
  }
#endif

  // tail (C % TILE elements)
  for (int e = numTiles * TILE + tid; e < C; e += WG) consume(rp[e], e);

  // ---- wave32 butterfly reduction ----
  const int lane = tid & 31;
  const int wave = tid >> 5;
#pragma unroll
  for (int off = 16; off >= 1; off >>= 1) {
    const float m2  = __shfl_xor(m, off, 32);
    const float s2  = __shfl_xor(s, off, 32);
    const float xy2 = __shfl_xor(xy, off, 32);
    float r[KTOP];
#pragma unroll
    for (int j = 0; j < KTOP; ++j) r[j] = __shfl_xor(t[j], off, 32);  // snapshot
    const float M = fmaxf(m, m2);
    s = s * __expf(m - M) + s2 * __expf(m2 - M);
    m = M;
    xy = fmaxf(xy, xy2);
#pragma unroll
    for (int j = 0; j < KTOP; ++j) top6_insert(t, r[j]);
  }

  if (lane == 0) {
    float* e = &red[wave * 9];
    e[0] = m; e[1] = s; e[2] = xy;
#pragma unroll
    for (int j = 0; j < KTOP; ++j) e[3 + j] = t[j];
  }
  __syncthreads();

  if (tid == 0) {
    float M = red[0], S = red[1], XY = red[2];
    float tt[KTOP];
#pragma unroll
    for (int j = 0; j < KTOP; ++j) tt[j] = red[3 + j];
    for (int w = 1; w < 8; ++w) {
      const float* e = &red[w * 9];
      const float m2 = e[0], s2 = e[1];
      const float Mn = fmaxf(M, m2);
      S = S * __expf(M - Mn) + s2 * __expf(m2 - Mn);
      M = Mn;
      XY = fmaxf(XY, e[2]);
#pragma unroll
      for (int j = 0; j < KTOP; ++j) top6_insert(tt, e[3 + j]);
    }
    const float pp = __expf(XY - M) / S;   // positive-class probability
    float acc = 0.0f;
#pragma unroll
    for (int j = 0; j < KTOP; ++j) {
      const float pj = __expf(tt[j] - M) / S;
      const float d = 1.0f + pj - pp;
      acc += d * d;
    }
    row_loss[row] = acc * (1.0f / 5.0f);   // / K
  }
}

__global__ __launch_bounds__(WG) void autkc_finalize_kernel(
    const float* __restrict__ row_loss, float* __restrict__ out, int B) {
  __shared__ float sm[WG];
  float acc = 0.0f;
  for (int i = threadIdx.x; i < B; i += WG) acc += row_loss[i];  // fixed order
  sm[threadIdx.x] = acc;
  __syncthreads();
  for (int s = WG / 2; s > 0; s >>= 1) {
    if (threadIdx.x < (unsigned)s) sm[threadIdx.x] += sm[threadIdx.x + s];
    __syncthreads();
  }
  if (threadIdx.x == 0) out[0] = sm[0] / (float)B;  // deterministic mean
}

extern "C" void kernel_launch(void* const* d_in, const int* in_sizes, int n_in,
                              void* d_out, int out_size, void* d_ws, size_t ws_size,
                              hipStream_t stream) {
  const float* pred = (const float*)d_in[0];
  const int* y = (const int*)d_in[1];
  // d_in[2] = epoch (unused: epoch >= epoch_to_paced always takes AUTKC branch)
  const int B = in_sizes[1];
  const int C = in_sizes[0] / B;
  float* row_loss = (float*)d_ws;   // B floats of scratch
  float* out = (float*)d_out;

  autkc_row_kernel<<<B, WG, 0, stream>>>(pred, y, row_loss, C);
  autkc_finalize_kernel<<<1, WG, 0, stream>>>(row_loss, out, B);
}